// EKFGradient_67594195304754
// MI455X (gfx1250) — compile-verified
//
#include <hip/hip_runtime.h>
#include <hip/hip_bf16.h>
#include <stdint.h>
#include <stddef.h>

#define GLOBAL_AS __attribute__((address_space(1)))
#define LDS_AS    __attribute__((address_space(3)))

#ifndef __has_builtin
#define __has_builtin(x) 0
#endif

typedef int v4i __attribute__((ext_vector_type(4)));

// ---------------- CDNA5 async global->LDS copy (16B per lane) ----------------
#if __has_builtin(__builtin_amdgcn_global_load_async_to_lds_b128)
#define HAVE_ASYNC_BUILTIN 1
#else
#define HAVE_ASYNC_BUILTIN 0
#endif

__device__ __forceinline__ void async_cp16(const void* gsrc, void* ldst) {
#if HAVE_ASYNC_BUILTIN
  __builtin_amdgcn_global_load_async_to_lds_b128(
      (GLOBAL_AS v4i*)gsrc, (LDS_AS v4i*)ldst, 0, 0);
#else
  unsigned       loff = (unsigned)(unsigned long long)(LDS_AS char*)ldst;
  unsigned long long ga = (unsigned long long)gsrc;
  asm volatile("global_load_async_to_lds_b128 %0, %1, off"
               :: "v"(loff), "v"(ga) : "memory");
#endif
}

#if __has_builtin(__builtin_amdgcn_s_wait_asynccnt)
#define S_WAIT_ASYNCCNT(n) do { __builtin_amdgcn_s_wait_asynccnt(n); \
                                asm volatile("" ::: "memory"); } while (0)
#else
#define S_WAIT_ASYNCCNT_STR(n) asm volatile("s_wait_asynccnt " #n ::: "memory")
#define S_WAIT_ASYNCCNT(n) S_WAIT_ASYNCCNT_STR(n)
#endif

#define S_WAIT_DS0() asm volatile("s_wait_dscnt 0x0" ::: "memory")

// ---------------- fast transcendentals ----------------
__device__ __forceinline__ float fast_rcp(float x) {
  return __builtin_amdgcn_rcpf(x);
}

__device__ __forceinline__ float fast_tanh(float x) {
#if __has_builtin(__builtin_amdgcn_tanhf)
  return __builtin_amdgcn_tanhf(x);
#elif __has_builtin(__builtin_amdgcn_tanh_f32)
  return __builtin_amdgcn_tanh_f32(x);
#else
  // tanh(x) = 1 - 2/(e^{2x}+1); saturates correctly for |x| large
  return 1.0f - 2.0f * fast_rcp(__expf(2.0f * x) + 1.0f);
#endif
}

// ---------------- problem constants ----------------
#define N_TRAJ      2048
#define T_STEPS     512
#define TS          32                 // steps per LDS tile
#define TILE_BYTES  (TS * 3 * 4)       // 384 B contiguous per lane per tile
#define LANE_STRIDE 400                // padded (16B aligned, odd multiple of 16)
#define BUF_BYTES   (32 * LANE_STRIDE) // 12800 B per wave buffer
#define WAVE_BYTES  (2 * BUF_BYTES)    // double buffered
#define DT_         (1.0f / 120.0f)

__global__ __launch_bounds__(64, 1) void
ekf_losses_kernel(const float* __restrict__ meas,       // [2048][512][3]
                  const float* __restrict__ init_state, // [2048][6]
                  const float* __restrict__ dyna,       // [4]
                  const float* __restrict__ Qm,         // [6][6] diagonal
                  const float* __restrict__ Rm,         // [3][3] diagonal
                  const float* __restrict__ P0,         // [6][6] diagonal
                  float* __restrict__ out)              // [2048][512]
{
  __shared__ __align__(16) char smem[2 * WAVE_BYTES];   // 51200 B: 2 waves

  const int tid  = (int)threadIdx.x;
  const int lane = tid & 31;
  const int wave = tid >> 5;
  const int traj = (int)blockIdx.x * 64 + tid;          // 32 blocks x 64 = 2048

  // uniform parameters (scalar loads)
  const float fric = dyna[0];
  const float damp = dyna[1];
  const float q0 = Qm[0],  q1 = Qm[7],  q2 = Qm[14];
  const float q3 = Qm[21], q4 = Qm[28], q5 = Qm[35];
  const float r0 = Rm[0],  r1 = Rm[4],  r2 = Rm[8];

  // per-trajectory state (registers for the whole kernel)
  float sx = init_state[traj * 6 + 0];
  float sy = init_state[traj * 6 + 1];
  float vx = init_state[traj * 6 + 2];
  float vy = init_state[traj * 6 + 3];
  float st = init_state[traj * 6 + 4];
  float vt = init_state[traj * 6 + 5];

  // P stays exactly block-diagonal (three 2x2 blocks) given diagonal Q/R/P0:
  // (a,b,c) = (P[pos,pos], P[pos,vel], P[vel,vel]) per subsystem.
  float xa = P0[0],  xb = P0[2],  xc = P0[14];
  float ya = P0[7],  yb = P0[9],  yc = P0[21];
  float ta = P0[28], tb = P0[29], tc = P0[35];

  const char* gbase = (const char*)(meas + (size_t)traj * (T_STEPS * 3));
  char* lane_buf0 = smem + wave * WAVE_BYTES + lane * LANE_STRIDE;
  char* lane_buf1 = lane_buf0 + BUF_BYTES;

  // prefetch tile 0 into buffer 0: 24 x b128 per lane (contiguous 384 B)
  {
#pragma unroll
    for (int i = 0; i < 24; ++i)
      async_cp16(gbase + i * 16, lane_buf0 + i * 16);
  }

  float* outp = out + (size_t)traj * T_STEPS;

  for (int tile = 0; tile < T_STEPS / TS; ++tile) {
    const int cur = tile & 1;

    if (tile + 1 < T_STEPS / TS) {
      // make sure pending LDS reads of the buffer we overwrite have retired
      S_WAIT_DS0();
      const char* gsrc = gbase + (size_t)(tile + 1) * TILE_BYTES;
      char* dst = cur ? lane_buf0 : lane_buf1;
#pragma unroll
      for (int i = 0; i < 24; ++i)
        async_cp16(gsrc + i * 16, dst + i * 16);
      // async loads complete in order: cnt<=24 => current tile's 24 are done
      S_WAIT_ASYNCCNT(24);
    } else {
      S_WAIT_ASYNCCNT(0);
    }

    const char* zb = cur ? lane_buf1 : lane_buf0;

#pragma unroll 4
    for (int tl = 0; tl < TS; ++tl) {
      const float* z = (const float*)(zb + tl * 12);
      const float z0 = z[0], z1 = z[1], z2 = z[2];

      // ================= x subsystem (state: sx, vx; P: xa,xb,xc) ==========
      float tx = fast_tanh(100.0f * vx);
      float ax = 1.0f - DT_ * (damp + 100.0f * fric * (1.0f - tx * tx));
      float px = sx + DT_ * vx;
      float wx = vx - DT_ * (damp * vx + fric * tx);
      float m0 = xa + DT_ * xb;            // (F P) row0
      float m1 = xb + DT_ * xc;
      float n0 = m0 + DT_ * m1 + q0;       // P_pred
      float n1 = ax * m1;
      float n2 = ax * (ax * xc) + q2;
      float S0 = n0 + r0;
      float is0 = fast_rcp(S0);
      float y0 = z0 - px;
      float k0 = n0 * is0, k1 = n1 * is0;  // Kalman gain
      sx = px + k0 * y0;  vx = wx + k1 * y0;
      xa = n0 - k0 * n0;  xb = n1 - k0 * n1;  xc = n2 - k1 * n1;

      // ================= y subsystem =======================================
      float ty = fast_tanh(100.0f * vy);
      float ay = 1.0f - DT_ * (damp + 100.0f * fric * (1.0f - ty * ty));
      float py = sy + DT_ * vy;
      float wy = vy - DT_ * (damp * vy + fric * ty);
      float u0 = ya + DT_ * yb;
      float u1 = yb + DT_ * yc;
      float o0 = u0 + DT_ * u1 + q1;
      float o1 = ay * u1;
      float o2 = ay * (ay * yc) + q3;
      float S1 = o0 + r1;
      float is1 = fast_rcp(S1);
      float y1 = z1 - py;
      float g0 = o0 * is1, g1 = o1 * is1;
      sy = py + g0 * y1;  vy = wy + g1 * y1;
      ya = o0 - g0 * o0;  yb = o1 - g0 * o1;  yc = o2 - g1 * o1;

      // ================= theta subsystem (linear, F=[[1,dt],[0,1]]) ========
      float pt = st + DT_ * vt;
      float e0 = ta + DT_ * tb;
      float e1 = tb + DT_ * tc;
      float f0 = e0 + DT_ * e1 + q4;
      float f1 = e1;
      float f2 = tc + q5;
      float S2 = f0 + r2;
      float is2 = fast_rcp(S2);
      float y2 = z2 - pt;
      float h0 = f0 * is2, h1 = f1 * is2;
      st = pt + h0 * y2;  vt = vt + h1 * y2;
      ta = f0 - h0 * f0;  tb = f1 - h0 * f1;  tc = f2 - h1 * f1;

      // loss: det(S) + y^T S^-1 y  (S exactly diagonal)
      float det  = S0 * S1 * S2;
      float maha = y0 * y0 * is0 + y1 * y1 * is1 + y2 * y2 * is2;
      outp[tile * TS + tl] = det + maha;
    }
  }
}

extern "C" void kernel_launch(void* const* d_in, const int* in_sizes, int n_in,
                              void* d_out, int out_size, void* d_ws, size_t ws_size,
                              hipStream_t stream) {
  (void)in_sizes; (void)n_in; (void)out_size; (void)d_ws; (void)ws_size;
  const float* meas = (const float*)d_in[0];
  const float* init = (const float*)d_in[1];
  const float* dyna = (const float*)d_in[2];
  const float* Qm   = (const float*)d_in[3];
  const float* Rm   = (const float*)d_in[4];
  const float* P0   = (const float*)d_in[5];
  float* out = (float*)d_out;

  ekf_losses_kernel<<<dim3(N_TRAJ / 64), dim3(64), 0, stream>>>(
      meas, init, dyna, Qm, Rm, P0, out);
}